// SignatureLayer_24996709662819
// MI455X (gfx1250) — compile-verified
//
#include <hip/hip_runtime.h>

typedef __attribute__((ext_vector_type(2))) float v2f;
typedef __attribute__((ext_vector_type(4))) float v4f;
typedef __attribute__((ext_vector_type(8))) float v8f;

namespace {
constexpr int CH   = 8;     // channels
constexpr int LEN  = 128;   // path length
constexpr int KDIM = 16;    // GEMM K (fully used, no padding)
constexpr int ASTR = 17;    // padded LDS row stride for A (bank-conflict free)
constexpr int OUT_STRIDE = 8 + 64 + 512 + 4096;  // 4680
}

// One wave (32 threads) per batch element. S4 (64x64, rows=ij, cols=kl) lives
// in 16 WMMA accumulator tiles. Per time step:
//   N4 = S4 + A(64x16) x B(16x64) with v_wmma_f32_16x16x4_f32 (D = A*B + C)
// where (algebraically folded, exact):
//   A[ij][k]   = S3[ijk] + 0.5*S2[ij]*d[k]   (covers S3(x)d + S2(x)d2)
//   B[k][k',l] = delta_{kk'} * d[l]
//   A[ij][8+m] = (S1[i] + 0.25*d[i]) * delta_{jm}
//   B[8+m][kl] = d3[m,k,l] = d[m]d[k]d[l]/6  (covers S1(x)d3 + d4)
__global__ __launch_bounds__(32) void sig4_wmma_kernel(
    const float* __restrict__ path, float* __restrict__ out)
{
    __shared__ float P[(LEN + 1) * CH];   // zero basepoint row + path
    __shared__ float S1[CH];
    __shared__ float S2[CH * CH];
    __shared__ float S3[CH * CH * CH];
    __shared__ float Am[64 * ASTR];       // [row(ij)][k], stride 17
    __shared__ float Bm[KDIM * 64];       // [k][col(kl)]

    const int lane = threadIdx.x;         // 0..31
    const int b    = blockIdx.x;
    const float* pb = path + (size_t)b * LEN * CH;

    // ---- stage full path into LDS (coalesced b128), basepoint row = 0 ----
    if (lane < CH) P[lane] = 0.f;
    {
        const v4f* src = (const v4f*)pb;
        v4f* dst = (v4f*)(P + CH);
        #pragma unroll
        for (int i = lane; i < LEN * CH / 4; i += 32) dst[i] = src[i];
    }
    // zero-init state + A/B (structured zeros must persist)
    for (int i = lane; i < CH; i += 32)        S1[i] = 0.f;
    for (int i = lane; i < CH * CH; i += 32)   S2[i] = 0.f;
    for (int i = lane; i < 512; i += 32)       S3[i] = 0.f;
    for (int i = lane; i < 64 * ASTR; i += 32) Am[i] = 0.f;
    for (int i = lane; i < KDIM * 64; i += 32) Bm[i] = 0.f;
    __syncthreads();

    v8f acc[4][4];
    #pragma unroll
    for (int ti = 0; ti < 4; ++ti)
        #pragma unroll
        for (int tj = 0; tj < 4; ++tj) {
            v8f z = {0.f, 0.f, 0.f, 0.f, 0.f, 0.f, 0.f, 0.f};
            acc[ti][tj] = z;
        }

    const int halfsel = lane >> 4;   // 0: lanes 0-15, 1: lanes 16-31
    const int lid     = lane & 15;

    #pragma unroll 1
    for (int l = 0; l < LEN; ++l) {
        // ---- increment d = p[l] - p[l-1], broadcast-read from LDS ----
        float d[CH];
        #pragma unroll
        for (int c = 0; c < CH; ++c)
            d[c] = P[(l + 1) * CH + c] - P[l * CH + c];

        // ---- build A (old S, folded) and B (exp(d) pieces) in LDS ----
        #pragma unroll
        for (int t = 0; t < 2; ++t) {
            int idx = lane * 2 + t;                 // 0..63
            int k = idx >> 3, lp = idx & 7;
            Bm[k * 64 + k * 8 + lp] = d[lp];        // B rows 0..7 block-diag
            int i = idx >> 3, j = idx & 7;
            Am[idx * ASTR + 8 + j] = S1[i] + 0.25f * d[i];  // A cols 8..15
        }
        #pragma unroll
        for (int t = 0; t < 16; ++t) {
            int idx = lane * 16 + t;                // 0..511
            int m = idx >> 6, k = (idx >> 3) & 7, lp = idx & 7;
            Bm[(8 + m) * 64 + k * 8 + lp] =
                d[m] * d[k] * d[lp] * (1.0f / 6.0f);        // B rows 8..15: d3
            int row = idx >> 3, kc = idx & 7;
            Am[row * ASTR + kc] = S3[idx] + 0.5f * S2[row] * d[kc];
        }
        __syncthreads();

        // ---- update S1,S2,S3 (read old, sync, write new) ----
        float n3[16];
        #pragma unroll
        for (int t = 0; t < 16; ++t) {
            int idx = lane * 16 + t;
            int i = idx >> 6, j = (idx >> 3) & 7, k = idx & 7;
            n3[t] = S3[idx] + S2[i * 8 + j] * d[k]
                  + S1[i] * (0.5f * d[j] * d[k])
                  + d[i] * d[j] * d[k] * (1.0f / 6.0f);
        }
        float n2[2];
        #pragma unroll
        for (int t = 0; t < 2; ++t) {
            int idx = lane * 2 + t;
            int i = idx >> 3, j = idx & 7;
            n2[t] = S2[idx] + S1[i] * d[j] + 0.5f * d[i] * d[j];
        }
        float n1 = (lane < CH) ? (S1[lane & 7] + d[lane & 7]) : 0.f;
        __syncthreads();
        #pragma unroll
        for (int t = 0; t < 16; ++t) S3[lane * 16 + t] = n3[t];
        #pragma unroll
        for (int t = 0; t < 2; ++t)  S2[lane * 2 + t]  = n2[t];
        if (lane < CH) S1[lane] = n1;

        // ---- S4 += A x B via WMMA f32 16x16x4: 4x4 tiles, 4 K-steps ----
        #pragma unroll
        for (int ti = 0; ti < 4; ++ti) {
            #pragma unroll
            for (int tj = 0; tj < 4; ++tj) {
                #pragma unroll
                for (int kk = 0; kk < KDIM / 4; ++kk) {
                    int ak   = kk * 4 + halfsel * 2;
                    int arow = ti * 16 + lid;
                    int bcol = tj * 16 + lid;
                    v2f av, bv;
                    av.x = Am[arow * ASTR + ak];
                    av.y = Am[arow * ASTR + ak + 1];
                    bv.x = Bm[ak * 64 + bcol];
                    bv.y = Bm[(ak + 1) * 64 + bcol];
                    acc[ti][tj] = __builtin_amdgcn_wmma_f32_16x16x4_f32(
                        false, av, false, bv, (short)0, acc[ti][tj],
                        false, false);
                }
            }
        }
        __syncthreads();   // protect A/B + S before next iteration rebuilds
    }

    // ---- write out: [S1 | S2 | S3 | S4] row-major per batch ----
    float* ob = out + (size_t)b * OUT_STRIDE;
    if (lane < CH) ob[lane] = S1[lane];
    #pragma unroll
    for (int t = 0; t < 2; ++t)  ob[8 + lane * 2 + t]   = S2[lane * 2 + t];
    #pragma unroll
    for (int t = 0; t < 16; ++t) ob[72 + lane * 16 + t] = S3[lane * 16 + t];
    float* o4 = ob + 72 + 512;
    #pragma unroll
    for (int ti = 0; ti < 4; ++ti)
        #pragma unroll
        for (int tj = 0; tj < 4; ++tj)
            #pragma unroll
            for (int r = 0; r < 8; ++r) {
                int row = ti * 16 + r + halfsel * 8;   // M index of VGPR r
                int col = tj * 16 + lid;               // N index = lane%16
                o4[row * 64 + col] = acc[ti][tj][r];
            }
}

extern "C" void kernel_launch(void* const* d_in, const int* in_sizes, int n_in,
                              void* d_out, int out_size, void* d_ws, size_t ws_size,
                              hipStream_t stream) {
    const float* path = (const float*)d_in[0];
    float* out = (float*)d_out;
    int B = in_sizes[0] / (LEN * CH);   // 4096
    sig4_wmma_kernel<<<B, 32, 0, stream>>>(path, out);
}